// QuantumModel_32375463478057
// MI455X (gfx1250) — compile-verified
//
#include <hip/hip_runtime.h>
#include <math.h>

#define NQ     10
#define DIM    1024          // 2^NQ
#define BATCHN 8192
#define NCLS   10
#define ROWS   32            // batch rows per workgroup (2 M-tiles per wave)
#define PAD    1028          // psi LDS row stride (floats)
#define PPAD   1025          // probs LDS row stride (floats), 1025 % 64 == 1

typedef __attribute__((ext_vector_type(2))) float v2f;
typedef __attribute__((ext_vector_type(8))) float v8f;

// ----------------------------------------------------------------------------
// Phase 1: build the circuit unitary, transposed, from all_params.
// Block k evolves basis state |k> in LDS; final amplitudes = row k of U^T.
// Output layout (K-pair interleaved for phase-2 B fragments):
//   ut float2[(k>>1)*DIM + j] = { U^T[2m][j], U^T[2m+1][j] }.
// ----------------------------------------------------------------------------

struct C2x2 { float r00,i00, r01,i01, r10,i10, r11,i11; };

__device__ __forceinline__ C2x2 mRX(float t){ float c=cosf(0.5f*t), s=sinf(0.5f*t);
  return C2x2{c,0.f, 0.f,-s, 0.f,-s, c,0.f}; }
__device__ __forceinline__ C2x2 mRY(float t){ float c=cosf(0.5f*t), s=sinf(0.5f*t);
  return C2x2{c,0.f, -s,0.f, s,0.f, c,0.f}; }
__device__ __forceinline__ C2x2 mRZ(float t){ float c=cosf(0.5f*t), s=sinf(0.5f*t);
  return C2x2{c,-s, 0.f,0.f, 0.f,0.f, c,s}; }

// single-qubit gate; wire w maps to bit position (NQ-1-w) of the flat index
__device__ __forceinline__ void apply1q(float* sr, float* si, int wire, C2x2 u) {
  const int p = (NQ - 1) - wire;
  const int s = 1 << p;
  #pragma unroll
  for (int rep = 0; rep < 2; ++rep) {
    int t  = threadIdx.x + rep * 256;                 // 512 pairs total
    int j0 = ((t >> p) << (p + 1)) | (t & (s - 1));
    int j1 = j0 | s;
    float ar = sr[j0], ai = si[j0];
    float br = sr[j1], bi = si[j1];
    sr[j0] = u.r00*ar - u.i00*ai + u.r01*br - u.i01*bi;
    si[j0] = u.r00*ai + u.i00*ar + u.r01*bi + u.i01*br;
    sr[j1] = u.r10*ar - u.i10*ai + u.r11*br - u.i11*bi;
    si[j1] = u.r10*ai + u.i10*ar + u.r11*bi + u.i11*br;
  }
  __syncthreads();
}

// insert two zero bits at positions lo < hi into an 8-bit value
__device__ __forceinline__ int expand2(int t, int lo, int hi) {
  int low  = t & ((1 << lo) - 1);
  int mid  = (t >> lo) & ((1 << (hi - lo - 1)) - 1);
  int high = t >> (hi - 1);
  return low | (mid << (lo + 1)) | (high << (hi + 1));
}

__device__ __forceinline__ void applyCZ(float* sr, float* si, int a, int b) {
  int pa = (NQ-1)-a, pb = (NQ-1)-b;
  int lo = pa < pb ? pa : pb, hi = pa < pb ? pb : pa;
  int j = expand2((int)threadIdx.x, lo, hi) | (1 << lo) | (1 << hi);
  sr[j] = -sr[j]; si[j] = -si[j];
  __syncthreads();
}

__device__ __forceinline__ void applyCNOT(float* sr, float* si, int c, int t) {
  int pc = (NQ-1)-c, pt = (NQ-1)-t;
  int lo = pc < pt ? pc : pt, hi = pc < pt ? pt : pc;
  int j0 = expand2((int)threadIdx.x, lo, hi) | (1 << pc);   // ctrl=1, tgt=0
  int j1 = j0 | (1 << pt);
  float tr = sr[j0], ti = si[j0];
  sr[j0] = sr[j1]; si[j0] = si[j1];
  sr[j1] = tr;     si[j1] = ti;
  __syncthreads();
}

__device__ void vqc_block(float* sr, float* si, const float* __restrict__ p, int i) {
  apply1q(sr,si,i,   mRX(p[0]));
  apply1q(sr,si,i+1, mRY(p[1]));
  apply1q(sr,si,i,   mRY(p[2]));
  apply1q(sr,si,i+1, mRZ(p[3]));
  apply1q(sr,si,i,   mRZ(p[4]));
  apply1q(sr,si,i+1, mRX(p[5]));
  applyCZ(sr,si,i,i+1);
  apply1q(sr,si,i,   mRZ(p[6]));
  apply1q(sr,si,i+1, mRX(p[7]));
  apply1q(sr,si,i,   mRY(p[8]));
  apply1q(sr,si,i+1, mRZ(p[9]));
  apply1q(sr,si,i,   mRX(p[10]));
  apply1q(sr,si,i+1, mRY(p[11]));
  applyCNOT(sr,si,i+1,i);
}

__global__ __launch_bounds__(256)
void build_unitary(const float* __restrict__ prm,
                   float* __restrict__ utr, float* __restrict__ uti) {
  __shared__ float sr[DIM];
  __shared__ float si[DIM];
  const int k = blockIdx.x;
  for (int i = threadIdx.x; i < DIM; i += 256) { sr[i] = (i == k) ? 1.f : 0.f; si[i] = 0.f; }
  __syncthreads();

  const float h = 0.70710678118654752440f;
  for (int w = 0; w < NQ; ++w) apply1q(sr,si,w, C2x2{h,0.f, h,0.f, h,0.f, -h,0.f});
  for (int w = 0; w < NQ; ++w) apply1q(sr,si,w, mRY(prm[w]));
  for (int b = 0; b < NQ/2; ++b) vqc_block(sr,si, prm + NQ      + 12*b, 2*b);
  for (int b = 0; b < NQ/2; ++b) vqc_block(sr,si, prm + NQ + 60 + 12*b, 2*b);
  for (int w = 0; w < NQ; ++w) applyCNOT(sr,si, w, (w+1) % NQ);
  for (int w = 0; w < NQ; ++w) {                     // rot = rz(om)@ry(th)@rz(ph)
    const float* sel = prm + NQ + 120 + 3*w;
    apply1q(sr,si,w, mRZ(sel[0]));
    apply1q(sr,si,w, mRY(sel[1]));
    apply1q(sr,si,w, mRZ(sel[2]));
  }
  for (int w = 0; w < NQ; ++w) applyCNOT(sr,si, w, (w+1) % NQ);

  // write row k of U^T, K-pair interleaved: float index (kp*DIM + j)*2 + (k&1)
  const int kp = k >> 1, klo = k & 1;
  for (int j = threadIdx.x; j < DIM; j += 256) {
    utr[(size_t)(kp * DIM + j) * 2 + klo] = sr[j];
    uti[(size_t)(kp * DIM + j) * 2 + klo] = si[j];
  }
}

// ----------------------------------------------------------------------------
// Phase 2: 32 batch rows per workgroup.
//   psi = x/||x|| staged in LDS. Each of 8 waves owns 8 column tiles; for each
//   column tile it computes TWO 16x16 output tiles (rows 0-15 and 16-31) that
//   share the same B fragments -> 4 WMMA per 2 global b64 loads, halving L2
//   traffic on U vs a 16-row tile. |psi_out|^2 tiles are staged to a probs
//   LDS plane; the sign-weighted class reduction, x_proj, and final FC are
//   cheap cooperative epilogues.
// ----------------------------------------------------------------------------

__global__ __launch_bounds__(256)
void batched_apply(const float* __restrict__ x,
                   const float2* __restrict__ utr, const float2* __restrict__ uti,
                   const float* __restrict__ w_in, const float* __restrict__ b_in,
                   const float* __restrict__ w_fc, const float* __restrict__ b_fc,
                   float* __restrict__ out) {
  __shared__ __align__(16) float psi[ROWS * PAD];    // ~131.5 KB
  __shared__ __align__(16) float probs[ROWS * PPAD]; // ~131.2 KB
  __shared__ float norm2[ROWS];
  __shared__ float normv[ROWS];
  __shared__ float q_lds[ROWS][NCLS];
  __shared__ float xp_lds[ROWS][NCLS];

  const int tid  = threadIdx.x;
  const int lane = tid & 31;
  const int wv   = tid >> 5;                  // wave 0..7
  const int b0   = blockIdx.x * ROWS;

  if (tid < ROWS) norm2[tid] = 0.f;
  __syncthreads();

  // ---- load x tile + row sum-of-squares; 8 threads per row, 128 cols each --
  {
    const int r  = tid >> 3;                  // 0..31
    const int c0 = (tid & 7) << 7;            // 0,128,...,896
    const float* xr = x + (size_t)(b0 + r) * DIM + c0;
    float* pr = psi + r * PAD + c0;
    float ss = 0.f;
    #pragma unroll 8
    for (int k = 0; k < 128; ++k) { float v = xr[k]; pr[k] = v; ss += v * v; }
    atomicAdd(&norm2[r], ss);
  }
  __syncthreads();
  {
    const int r  = tid >> 3;
    const int c0 = (tid & 7) << 7;
    const float inv = rsqrtf(norm2[r]);
    float* pr = psi + r * PAD + c0;
    #pragma unroll 8
    for (int k = 0; k < 128; ++k) pr[k] *= inv;
    if ((tid & 7) == 0) normv[r] = sqrtf(norm2[r]);
  }
  __syncthreads();

  // ---- WMMA main loop: 2 M-tiles share each B fragment ----
  const int kpoff = lane >> 4;                         // 0: K0/K1, 1: K2/K3
  const float* aptr0 = psi + (lane & 15) * PAD + (kpoff << 1);       // rows 0-15
  const float* aptr1 = aptr0 + 16 * PAD;                             // rows 16-31

  for (int t = 0; t < 8; ++t) {
    const int jt  = wv * 8 + t;                        // column tile 0..63
    const int col = jt * 16 + (lane & 15);             // absolute output column
    const float2* brp = utr + col;
    const float2* bip = uti + col;

    v8f accr0 = {}; v8f acci0 = {};
    v8f accr1 = {}; v8f acci1 = {};
    #pragma unroll 4
    for (int k0 = 0; k0 < DIM; k0 += 4) {
      const int kp = (k0 >> 1) + kpoff;
      float2 a0v = *(const float2*)(aptr0 + k0);
      float2 a1v = *(const float2*)(aptr1 + k0);
      float2 brv = brp[(size_t)kp * DIM];
      float2 biv = bip[(size_t)kp * DIM];
      v2f a0 = {a0v.x, a0v.y};
      v2f a1 = {a1v.x, a1v.y};
      v2f br = {brv.x, brv.y};
      v2f bi = {biv.x, biv.y};
      accr0 = __builtin_amdgcn_wmma_f32_16x16x4_f32(false, a0, false, br, (short)0, accr0, false, false);
      acci0 = __builtin_amdgcn_wmma_f32_16x16x4_f32(false, a0, false, bi, (short)0, acci0, false, false);
      accr1 = __builtin_amdgcn_wmma_f32_16x16x4_f32(false, a1, false, br, (short)0, accr1, false, false);
      acci1 = __builtin_amdgcn_wmma_f32_16x16x4_f32(false, a1, false, bi, (short)0, acci1, false, false);
    }
    // |psi_out|^2 -> probs LDS (disjoint columns per wave; no race)
    const int rhalf = (lane >> 4) << 3;                // 0 or 8
    #pragma unroll
    for (int v = 0; v < 8; ++v) {
      float p0 = accr0[v] * accr0[v] + acci0[v] * acci0[v];
      float p1 = accr1[v] * accr1[v] + acci1[v] * acci1[v];
      probs[(v + rhalf)      * PPAD + col] = p0;
      probs[(v + rhalf + 16) * PPAD + col] = p1;
    }
  }
  __syncthreads();

  // ---- q[r][c] = sum_j probs[r][j] * (1 - 2*bit_{c}(j)) ; 320 pairs ----
  #pragma unroll
  for (int rep = 0; rep < 2; ++rep) {
    const int p = tid + rep * 256;
    if (p < ROWS * NCLS) {
      const int r = p / NCLS, c = p % NCLS;
      const int sh = (NQ - 1) - c;
      const float* pr = probs + r * PPAD;
      float acc = 0.f;
      #pragma unroll 8
      for (int j = 0; j < DIM; ++j) {
        float v = pr[j];
        acc += ((j >> sh) & 1) ? -v : v;
      }
      q_lds[r][c] = acc;
    }
  }

  // ---- x_proj[r][c] = normv[r] * <psi_r, w_in_c> + b_in[c] ----
  #pragma unroll
  for (int rep = 0; rep < 2; ++rep) {
    const int p = tid + rep * 256;
    if (p < ROWS * NCLS) {
      const int r = p / NCLS, c = p % NCLS;
      const float* wr = w_in + (size_t)c * DIM;
      const float* pr = psi + r * PAD;
      float acc = 0.f;
      #pragma unroll 8
      for (int k = 0; k < DIM; ++k) acc += pr[k] * wr[k];
      xp_lds[r][c] = normv[r] * acc + b_in[c];
    }
  }
  __syncthreads();

  // ---- final FC: out = (q + x_proj) @ w_fc.T + b_fc ----
  #pragma unroll
  for (int rep = 0; rep < 2; ++rep) {
    const int p = tid + rep * 256;
    if (p < ROWS * NCLS) {
      const int r = p / NCLS, m = p % NCLS;
      float o = b_fc[m];
      #pragma unroll
      for (int c = 0; c < NCLS; ++c)
        o += w_fc[m * NCLS + c] * (q_lds[r][c] + xp_lds[r][c]);
      out[(size_t)(b0 + r) * NCLS + m] = o;
    }
  }
}

// ----------------------------------------------------------------------------

extern "C" void kernel_launch(void* const* d_in, const int* in_sizes, int n_in,
                              void* d_out, int out_size, void* d_ws, size_t ws_size,
                              hipStream_t stream) {
  const float* x      = (const float*)d_in[0];
  const float* params = (const float*)d_in[1];
  const float* w_in   = (const float*)d_in[2];
  const float* b_in   = (const float*)d_in[3];
  const float* w_fc   = (const float*)d_in[4];
  const float* b_fc   = (const float*)d_in[5];
  float* out = (float*)d_out;

  // workspace: UTr (4 MB) | UTi (4 MB), both K-pair-interleaved float2 planes
  float* utr = (float*)d_ws;
  float* uti = (float*)((char*)d_ws + (size_t)DIM * DIM * sizeof(float));

  build_unitary<<<DIM, 256, 0, stream>>>(params, utr, uti);
  batched_apply<<<BATCHN / ROWS, 256, 0, stream>>>(
      x, (const float2*)utr, (const float2*)uti, w_in, b_in, w_fc, b_fc, out);
}